// YOLOLayer_40261023433332
// MI455X (gfx1250) — compile-verified
//
#include <hip/hip_runtime.h>
#include <math.h>

// ---- problem geometry (RESO=416, gs=13, 3 anchors, 80 classes) ----
#define BLOCK      128
#define NUM_ATTRS  85          // 5 + 80
#define NCLS       80
#define GS         13
#define CELLS      169         // 13*13
#define CPI        507         // 3*169 cells per image
#define CH         255         // 3*85 channels
#define STRIDE_F   32.0f       // 416/13

// ---- gfx1250 async LDS<->global DMA (guarded so compile never breaks) ----
#if defined(__has_builtin)
#if __has_builtin(__builtin_amdgcn_global_store_async_from_lds_b128)
#define HAVE_ASYNC_STORE 1
#else
#define HAVE_ASYNC_STORE 0
#endif
#if __has_builtin(__builtin_amdgcn_s_wait_asynccnt)
#define HAVE_WAIT_ASYNC 1
#else
#define HAVE_WAIT_ASYNC 0
#endif
#else
#define HAVE_ASYNC_STORE 0
#define HAVE_WAIT_ASYNC 0
#endif

// Exact pointee type from the hipcc diagnostic:
//   param 1: 'int __attribute__((vector_size(16))) __device__ *'  (addrspace(1))
//   param 2: same vector type in LDS (addrspace(3))
typedef int v4i_t __attribute__((vector_size(4 * sizeof(int))));
typedef __attribute__((address_space(1))) v4i_t gv4i;   // global b128
typedef __attribute__((address_space(3))) v4i_t lv4i;   // LDS b128

__device__ __forceinline__ float fast_sigmoid(float v) {
    return 1.0f / (1.0f + __expf(-v));
}

__global__ void __launch_bounds__(BLOCK)
yolo_decode_kernel(const float* __restrict__ x, float* __restrict__ out,
                   int totalCells) {
    __shared__ float smem[BLOCK * NUM_ATTRS];   // 128*85*4 = 43,520 B

    const int tid       = threadIdx.x;
    const int blockBase = blockIdx.x * BLOCK;
    const int g         = blockBase + tid;      // flat cell id: b*507 + a*169 + r

    if (g < totalCells) {
        const int b    = g / CPI;
        const int cell = g - b * CPI;
        const int a    = cell / CELLS;
        const int r    = cell - a * CELLS;      // gy*13 + gx
        const int gy   = r / GS;
        const int gx   = r - gy * GS;

        // x[b, a*85 + k, gy, gx] ; consecutive lanes -> consecutive r -> coalesced
        const float* pin = x + ((size_t)b * CH + (size_t)a * NUM_ATTRS) * CELLS + r;

        float v[NUM_ATTRS];
#pragma unroll
        for (int k = 0; k < NUM_ATTRS; ++k) v[k] = pin[k * CELLS];

        float* row = smem + tid * NUM_ATTRS;    // stride 85 words: gcd(85%64,64)=1 -> no bank conflicts

        // scaled_anchor * stride == raw anchor (the /32 and *32 cancel)
        const float aw = (a == 0) ? 116.0f : ((a == 1) ? 156.0f : 373.0f);
        const float ah = (a == 0) ?  90.0f : ((a == 1) ? 198.0f : 326.0f);

        row[0] = (fast_sigmoid(v[0]) + (float)gx) * STRIDE_F;  // bx
        row[1] = (fast_sigmoid(v[1]) + (float)gy) * STRIDE_F;  // by
        row[2] = __expf(v[2]) * aw;                            // bw
        row[3] = __expf(v[3]) * ah;                            // bh
        row[4] = fast_sigmoid(v[4]);                           // conf

        // softmax over the 80 class logits (max-subtracted, like jax.nn.softmax)
        float m = v[5];
#pragma unroll
        for (int k = 6; k < NUM_ATTRS; ++k) m = fmaxf(m, v[k]);
        float s = 0.0f;
#pragma unroll
        for (int k = 0; k < NCLS; ++k) {
            const float e = __expf(v[5 + k] - m);
            s += e;
            row[5 + k] = e;                     // park in LDS, scale below (caps VGPRs)
        }
        const float inv = 1.0f / s;
#pragma unroll
        for (int k = 0; k < NCLS; ++k) row[5 + k] *= inv;
    }

    __syncthreads();

    // Block's output span is contiguous: out[blockBase*85 .. +cellsHere*85)
    int cellsHere = totalCells - blockBase;
    if (cellsHere > BLOCK) cellsHere = BLOCK;
    const int nwords = cellsHere * NUM_ATTRS;   // floats to copy
    const int nvec   = nwords >> 2;             // 16B chunks (block base is 16B aligned)
    float* gbase = out + (size_t)blockBase * NUM_ATTRS;

#if HAVE_ASYNC_STORE
    // CDNA5 async DMA: LDS -> global, b128 per lane, tracked by ASYNCcnt.
    for (int c = tid; c < nvec; c += BLOCK) {
        __builtin_amdgcn_global_store_async_from_lds_b128(
            (gv4i*)(gbase + (size_t)(c << 2)),
            (lv4i*)(smem + (c << 2)),
            0, 0);
    }
#else
    for (int c = tid; c < nvec; c += BLOCK) {
        const float4 val = *reinterpret_cast<const float4*>(smem + (c << 2));
        *reinterpret_cast<float4*>(gbase + (size_t)(c << 2)) = val;
    }
#endif
    // tail (never taken for bs*507 % 4 == 0, kept for generality)
    for (int w = (nvec << 2) + tid; w < nwords; w += BLOCK) gbase[w] = smem[w];

#if HAVE_ASYNC_STORE && HAVE_WAIT_ASYNC
    __builtin_amdgcn_s_wait_asynccnt(0);
#endif
}

extern "C" void kernel_launch(void* const* d_in, const int* in_sizes, int n_in,
                              void* d_out, int out_size, void* d_ws, size_t ws_size,
                              hipStream_t stream) {
    const float* x = (const float*)d_in[0];
    float* out = (float*)d_out;

    // bs = elems / (255*169); totalCells = bs * 507
    const long long elems = (long long)in_sizes[0];
    const int bs = (int)(elems / ((long long)CH * CELLS));
    const int totalCells = bs * CPI;
    const int blocks = (totalCells + BLOCK - 1) / BLOCK;   // 2028 for bs=512 (exact)

    yolo_decode_kernel<<<blocks, BLOCK, 0, stream>>>(x, out, totalCells);
}